// GCARGCN_369367188073
// MI455X (gfx1250) — compile-verified
//
#include <hip/hip_runtime.h>
#include <stdint.h>

#define NODES 50000
#define HD    128
#define RREL  8
#define BB    4
#define INDIM 384

typedef float v2f __attribute__((ext_vector_type(2)));
typedef float v8f __attribute__((ext_vector_type(8)));

__global__ void zero_f32(float* __restrict__ p, int n) {
    int i = blockIdx.x * blockDim.x + threadIdx.x;
    if (i < n) p[i] = 0.0f;
}

__global__ void degree_kernel(const int* __restrict__ dst, float* __restrict__ deg, int E) {
    int e = blockIdx.x * blockDim.x + threadIdx.x;
    if (e < E) atomicAdd(&deg[dst[e]], 1.0f);
}

// W_all[r,i,o] = sum_b comp[r,b] * V[b,i,o]
__global__ void combine_basis(const float* __restrict__ V, const float* __restrict__ comp,
                              float* __restrict__ W) {
    int idx = blockIdx.x * blockDim.x + threadIdx.x;
    if (idx >= RREL * HD * HD) return;
    int r  = idx >> 14;            // / (128*128)
    int io = idx & (HD * HD - 1);
    float s = 0.0f;
#pragma unroll
    for (int b = 0; b < BB; ++b)
        s = fmaf(comp[r * BB + b], V[b * HD * HD + io], s);
    W[idx] = s;
}

// C[M x 128] = A[M x K] @ W[K x 128] (+ bias, optional).
// One block: 16 output rows x 128 cols. 128 threads = 4 wave32; wave w owns
// column tiles [32w,32w+16) and [32w+16,32w+32). K multiple of 4; M multiple of 16.
// V_WMMA_F32_16X16X4_F32 lane mapping (32-bit matrices, wave32):
//   A 16x4 : lane = g*16+ln holds row M=ln, VGPR v holds K = k0 + v + 2g
//   B 4x16 : lane holds col N=ln,           VGPR v holds K = k0 + v + 2g
//   C 16x16: lane holds col N=ln,           VGPR v holds row M = v + 8g
// A tile is staged global->LDS with CDNA5 async copies (ASYNCcnt), not via VGPRs.
__global__ void gemm_wmma_f32(const float* __restrict__ A, const float* __restrict__ W,
                              const float* __restrict__ bias, float* __restrict__ C,
                              int K) {
    extern __shared__ float Alds[];          // 16 x (K+4), padded vs 64-bank LDS
    const int Kp   = K + 4;
    const int tid  = threadIdx.x;
    const int wave = tid >> 5;
    const int lane = tid & 31;
    const int g    = lane >> 4;
    const int ln   = lane & 15;
    const long rowBase = (long)blockIdx.x * 16;

#if defined(__gfx1250__)
    // Async global->LDS staging: one float4 per lane per issue, direct to LDS.
    // 16*(K/4) float4s; K=128 -> 4 issues/lane, K=384 -> 12; both exact multiples
    // of 128 threads so EXEC is all-ones throughout.
    {
        const unsigned ldsBase = (unsigned)(uintptr_t)Alds;  // addrspace(3) byte offset
        const int vecPerRow = K >> 2;
        for (int idx = tid; idx < 16 * vecPerRow; idx += 128) {
            int r = idx / vecPerRow;
            int c = idx - r * vecPerRow;                     // float4 index within row
            unsigned ldsAddr = ldsBase + (unsigned)(((r * Kp) + (c << 2)) << 2);
            const float* gp = A + (rowBase + r) * (long)K + (c << 2);
            asm volatile("global_load_async_to_lds_b128 %0, %1, off"
                         :: "v"(ldsAddr), "v"(gp)
                         : "memory");
        }
        asm volatile("s_wait_asynccnt 0x0" ::: "memory");
    }
#else
    for (int idx = tid; idx < 16 * K; idx += 128) {
        int r = idx / K;
        int c = idx - r * K;
        Alds[r * Kp + c] = A[(rowBase + r) * K + c];
    }
#endif
    __syncthreads();

    const int col0 = wave * 32 + ln;
    const int col1 = col0 + 16;

    v8f acc0 = {};
    v8f acc1 = {};
    const float* arow = Alds + ln * Kp;

    for (int k0 = 0; k0 < K; k0 += 4) {
        v2f a;
        a.x = arow[k0 + 2 * g];
        a.y = arow[k0 + 2 * g + 1];
        const float* w0 = W + (size_t)(k0 + 2 * g) * HD;   // row k0+2g
        v2f b0, b1;
        b0.x = w0[col0];      b0.y = w0[HD + col0];        // rows k0+2g, k0+2g+1
        b1.x = w0[col1];      b1.y = w0[HD + col1];
        acc0 = __builtin_amdgcn_wmma_f32_16x16x4_f32(false, a, false, b0,
                                                     (short)0, acc0, false, false);
        acc1 = __builtin_amdgcn_wmma_f32_16x16x4_f32(false, a, false, b1,
                                                     (short)0, acc1, false, false);
    }

    const float bv0 = bias ? bias[col0] : 0.0f;
    const float bv1 = bias ? bias[col1] : 0.0f;
#pragma unroll
    for (int v = 0; v < 8; ++v) {
        long row = rowBase + v + 8 * g;
        C[row * HD + col0] = acc0[v] + bv0;
        C[row * HD + col1] = acc1[v] + bv1;
    }
}

// One wave per edge. For edges of relation `rel`:
//   out[dst] += msg[src] * (1/max(deg[dst],1))
__global__ void edge_scatter(const float* __restrict__ msg, const int* __restrict__ src,
                             const int* __restrict__ dst, const int* __restrict__ etype,
                             const float* __restrict__ deg, float* __restrict__ out,
                             int E, int rel) {
    int e = blockIdx.x * 4 + (threadIdx.x >> 5);
    if (e >= E) return;
    if (etype[e] != rel) return;                  // wave-uniform predicate
    const int lane = threadIdx.x & 31;
    const int s = src[e];
    const int d = dst[e];
    const float nrm = 1.0f / fmaxf(deg[d], 1.0f);
    const float4 v = ((const float4*)(msg + (size_t)s * HD))[lane];
    float* drow = out + (size_t)d * HD + lane * 4;
    atomicAdd(drow + 0, v.x * nrm);
    atomicAdd(drow + 1, v.y * nrm);
    atomicAdd(drow + 2, v.z * nrm);
    atomicAdd(drow + 3, v.w * nrm);
}

__global__ void relu_kernel(float* __restrict__ p, long n) {
    long i = (long)blockIdx.x * blockDim.x + threadIdx.x;
    if (i < n) p[i] = fmaxf(p[i], 0.0f);
}

extern "C" void kernel_launch(void* const* d_in, const int* in_sizes, int n_in,
                              void* d_out, int out_size, void* d_ws, size_t ws_size,
                              hipStream_t stream) {
    (void)n_in; (void)out_size; (void)ws_size;

    const float* node_feats = (const float*)d_in[0];
    const int*   src        = (const int*)  d_in[1];
    const int*   dst        = (const int*)  d_in[2];
    const int*   etype      = (const int*)  d_in[3];
    const float* in_proj_W  = (const float*)d_in[4];
    const float* in_proj_b  = (const float*)d_in[5];
    const float* V1         = (const float*)d_in[6];
    const float* comp1      = (const float*)d_in[7];
    const float* loop1      = (const float*)d_in[8];
    const float* bias1      = (const float*)d_in[9];
    const float* V2         = (const float*)d_in[10];
    const float* comp2      = (const float*)d_in[11];
    const float* loop2      = (const float*)d_in[12];
    const float* bias2      = (const float*)d_in[13];
    float* out = (float*)d_out;
    const int E = in_sizes[1];

    // Workspace layout (floats); all offsets 16B-aligned. Total ~78 MB.
    float* ws   = (float*)d_ws;
    float* deg  = ws;                                 // N
    float* h0   = ws  + NODES;                        // N*128
    float* tmp  = h0  + (size_t)NODES * HD;           // N*128
    float* h1   = tmp + (size_t)NODES * HD;           // N*128
    float* Wall = h1  + (size_t)NODES * HD;           // 8*128*128

    const int mblocks  = NODES / 16;                  // 3125 (exact)
    const size_t ldsIn = (size_t)16 * (INDIM + 4) * sizeof(float);
    const size_t ldsH  = (size_t)16 * (HD + 4) * sizeof(float);

    // In-degree -> per-edge norm denominator
    zero_f32<<<(NODES + 255) / 256, 256, 0, stream>>>(deg, NODES);
    degree_kernel<<<(E + 255) / 256, 256, 0, stream>>>(dst, deg, E);

    // h0 = node_feats @ in_proj_W + in_proj_b
    gemm_wmma_f32<<<mblocks, 128, ldsIn, stream>>>(node_feats, in_proj_W, in_proj_b, h0, INDIM);

    // ---- Layer 1 ----
    combine_basis<<<(RREL * HD * HD + 255) / 256, 256, 0, stream>>>(V1, comp1, Wall);
    // base: h1 = h0 @ loop1 + bias1
    gemm_wmma_f32<<<mblocks, 128, ldsH, stream>>>(h0, loop1, bias1, h1, HD);
    for (int r = 0; r < RREL; ++r) {
        gemm_wmma_f32<<<mblocks, 128, ldsH, stream>>>(h0, Wall + (size_t)r * HD * HD,
                                                      nullptr, tmp, HD);
        edge_scatter<<<(E + 3) / 4, 128, 0, stream>>>(tmp, src, dst, etype, deg, h1, E, r);
    }
    relu_kernel<<<((long)NODES * HD + 255) / 256, 256, 0, stream>>>(h1, (long)NODES * HD);

    // ---- Layer 2 (writes straight into d_out) ----
    combine_basis<<<(RREL * HD * HD + 255) / 256, 256, 0, stream>>>(V2, comp2, Wall);
    gemm_wmma_f32<<<mblocks, 128, ldsH, stream>>>(h1, loop2, bias2, out, HD);
    for (int r = 0; r < RREL; ++r) {
        gemm_wmma_f32<<<mblocks, 128, ldsH, stream>>>(h1, Wall + (size_t)r * HD * HD,
                                                      nullptr, tmp, HD);
        edge_scatter<<<(E + 3) / 4, 128, 0, stream>>>(tmp, src, dst, etype, deg, out, E, r);
    }
}